// ConvolveTensor3body_76510547411073
// MI455X (gfx1250) — compile-verified
//
#include <hip/hip_runtime.h>
#include <hip/hip_bf16.h>
#include <math.h>

#define NODES 20000
#define NCH   128
#define DIM   16

typedef __attribute__((ext_vector_type(2))) float v2f;
typedef __attribute__((ext_vector_type(4))) float v4f;
typedef __attribute__((ext_vector_type(8))) float v8f;

// ---------------- d_ws layout (4-byte units) ----------------
// [0,4096)        dense W3J [i][j][k]
// [4096,4160)     per-(l1,l2,l3) inverse Frobenius norms (64)
// [4160,4192)     offs[17] (int) : CSR-by-k offsets into entry list
// [4192,8288)     entIJ[4096] (int)  : (i<<4)|j
// [8288,12384)    entW [4096] (float)
#define WS_INVN  4096
#define WS_OFFS  4160
#define WS_ENTIJ 4192
#define WS_ENTW  8288
#define WS_FLOATS_NEEDED 12384

struct cplx { double x, y; };
__device__ inline cplx cmul(cplx a, cplx b) {
  cplx r; r.x = a.x*b.x - a.y*b.y; r.y = a.x*b.y + a.y*b.x; return r;
}

__device__ inline double dfact(int n) {
  double r = 1.0;
  for (int i = 2; i <= n; ++i) r *= (double)i;
  return r;
}

__device__ double su2_cg(int j1, int m1, int j2, int m2, int j3, int m3) {
  if (m3 != m1 + m2) return 0.0;
  int vmin = -j1 + j2 + m3; if (-j1 + m1 > vmin) vmin = -j1 + m1; if (0 > vmin) vmin = 0;
  int vmax = j2 + j3 + m1; if (j3 - j1 + j2 < vmax) vmax = j3 - j1 + j2; if (j3 + m3 < vmax) vmax = j3 + m3;
  if (vmax < vmin) return 0.0;
  double C = sqrt((2.0*j3 + 1.0)
                  * dfact(j3 + j1 - j2) * dfact(j3 - j1 + j2) * dfact(j1 + j2 - j3)
                  * dfact(j3 + m3) * dfact(j3 - m3)
                  / (dfact(j1 + j2 + j3 + 1) * dfact(j1 - m1) * dfact(j1 + m1)
                     * dfact(j2 - m2) * dfact(j2 + m2)));
  double S = 0.0;
  for (int v = vmin; v <= vmax; ++v) {
    double sgn = ((v + j2 + m2) & 1) ? -1.0 : 1.0;
    S += sgn * dfact(j2 + j3 + m1 - v) * dfact(j1 - m1 + v)
         / (dfact(v) * dfact(j3 - j1 + j2 - v) * dfact(j3 + m3 - v) * dfact(v + j1 - j2 - m3));
  }
  return C * S;
}

__device__ inline int lval(int d) { return d >= 9 ? 3 : (d >= 4 ? 2 : (d >= 1 ? 1 : 0)); }

// Row r (0..2l) of the real->complex SH change-of-basis matrix, incl. (-i)^l phase.
// Each row has at most 2 nonzero columns.
__device__ int qrow(int l, int r, int cols[2], cplx vals[2]) {
  const double is2 = 0.70710678118654752440;
  int m = r - l, n;
  cplx v0, v1; int c0 = 0, c1 = 0;
  v1.x = 0.0; v1.y = 0.0;
  if (m < 0) {
    c0 = l - m; v0.x = is2;  v0.y = 0.0;      // q[l+m, l+|m|] = 1/sqrt2
    c1 = l + m; v1.x = 0.0;  v1.y = -is2;     // q[l+m, l-|m|] = -i/sqrt2
    n = 2;
  } else if (m == 0) {
    c0 = l;     v0.x = 1.0;  v0.y = 0.0;
    n = 1;
  } else {
    double s = (m & 1) ? -1.0 : 1.0;
    c0 = l + m; v0.x = s*is2; v0.y = 0.0;     // (-1)^m / sqrt2
    c1 = l - m; v1.x = 0.0;   v1.y = s*is2;   // i*(-1)^m / sqrt2
    n = 2;
  }
  cplx ph;
  switch (l & 3) {                            // (-i)^l
    case 0:  ph.x = 1.0;  ph.y = 0.0;  break;
    case 1:  ph.x = 0.0;  ph.y = -1.0; break;
    case 2:  ph.x = -1.0; ph.y = 0.0;  break;
    default: ph.x = 0.0;  ph.y = 1.0;  break;
  }
  vals[0] = cmul(v0, ph); cols[0] = c0;
  if (n == 2) { vals[1] = cmul(v1, ph); cols[1] = c1; }
  return n;
}

// ---------- setup kernel 1: dense real Wigner-3j (unnormalized) ----------
__global__ void build_w3j_dense(float* ws) {
  int id = blockIdx.x * blockDim.x + threadIdx.x;
  if (id >= 4096) return;
  int i = id >> 8, j = (id >> 4) & 15, k = id & 15;
  int l1 = lval(i), l2 = lval(j), l3 = lval(k);
  int lo = (l1 > l2) ? (l1 - l2) : (l2 - l1);
  float outv = 0.0f;
  if (l3 >= lo && l3 <= l1 + l2 && (((l1 + l2 + l3) & 1) == 0)) {
    int ac[2], bc[2], cc[2]; cplx av[2], bv[2], cv[2];
    int an = qrow(l1, i - l1*l1, ac, av);
    int bn = qrow(l2, j - l2*l2, bc, bv);
    int cn = qrow(l3, k - l3*l3, cc, cv);
    double re = 0.0;
    for (int a = 0; a < an; ++a)
      for (int b = 0; b < bn; ++b)
        for (int c = 0; c < cn; ++c) {
          double cg = su2_cg(l1, ac[a]-l1, l2, bc[b]-l2, l3, cc[c]-l3);
          if (cg == 0.0) continue;
          cplx q3c; q3c.x = cv[c].x; q3c.y = -cv[c].y;   // conj
          cplx p = cmul(cmul(av[a], bv[b]), q3c);
          re += p.x * cg;
        }
    outv = (float)re;
  }
  ws[(i*16 + j)*16 + k] = outv;
}

// ---------- setup kernel 2: per-block inverse Frobenius norms ----------
__global__ void w3j_norms(float* ws) {
  int t = threadIdx.x;
  if (t >= 64) return;
  int l1 = t >> 4, l2 = (t >> 2) & 3, l3 = t & 3;
  int lo = (l1 > l2) ? (l1 - l2) : (l2 - l1);
  float inv = 0.0f;
  if (l3 >= lo && l3 <= l1 + l2 && (((l1 + l2 + l3) & 1) == 0)) {
    double s = 0.0;
    for (int i = l1*l1; i < (l1+1)*(l1+1); ++i)
      for (int j = l2*l2; j < (l2+1)*(l2+1); ++j)
        for (int k = l3*l3; k < (l3+1)*(l3+1); ++k) {
          double v = (double)ws[(i*16 + j)*16 + k];
          s += v*v;
        }
    if (s > 0.0) inv = (float)(1.0 / sqrt(s));
  }
  ws[WS_INVN + t] = inv;
}

// ---------- setup kernel 3: pack normalized sparse list, sorted by k ----------
__global__ void w3j_pack(float* ws) {
  if (threadIdx.x != 0 || blockIdx.x != 0) return;
  int* offs   = (int*)(ws + WS_OFFS);
  int* entIJ  = (int*)(ws + WS_ENTIJ);
  float* entW = ws + WS_ENTW;
  int cnt = 0;
  for (int k = 0; k < 16; ++k) {
    offs[k] = cnt;
    int l3 = lval(k);
    for (int i = 0; i < 16; ++i) {
      int l1 = lval(i);
      for (int j = 0; j < 16; ++j) {
        int l2 = lval(j);
        float inv = ws[WS_INVN + l1*16 + l2*4 + l3];
        float v = ws[(i*16 + j)*16 + k] * inv;
        if (fabsf(v) > 1e-7f) { entIJ[cnt] = (i << 4) | j; entW[cnt] = v; ++cnt; }
      }
    }
  }
  offs[16] = cnt;
}

// SiLU with hardware fast reciprocal (v_rcp_f32, ~1 ulp)
__device__ inline float silu_fast(float y) {
  return y * __builtin_amdgcn_rcpf(1.0f + __expf(-y));
}

// ---------- main kernel: one node per 256-thread block (8 waves) ----------
__global__ void __launch_bounds__(256, 1) tp3_main(
    const float* __restrict__ x,  const float* __restrict__ W0,
    const float* __restrict__ W1, const float* __restrict__ W2,
    float* __restrict__ out,      const int* __restrict__ offs,
    const int* __restrict__ entIJ, const float* __restrict__ entW) {
  // 24KB output stage in exact global layout [c][k][3]; its first 8KB doubles
  // as the X stage (X is dead before the first OUTbuf write).
  __shared__ __align__(16) float OUTbuf[NCH*DIM*3];
  __shared__ float A1 [DIM*132];   // node1  in [i][c] (stride 132 vs banks)
  __shared__ float A1b[DIM*132];   // node1_
  __shared__ float A2b[DIM*132];   // node2_
  __shared__ float B2 [DIM*132];   // node2
  float* Xs = OUTbuf;              // [c][i], 2048 floats

  const int n = blockIdx.x;
  const int t = threadIdx.x;

  // -- stage node features (128x16 f32 = 8KB) --
  {
    const v4f* src = (const v4f*)(x + (size_t)n * (NCH*DIM));
    v4f* dst = (v4f*)Xs;
    v4f a = __builtin_nontemporal_load(src + t*2 + 0);
    v4f b = __builtin_nontemporal_load(src + t*2 + 1);
    dst[t*2 + 0] = a; dst[t*2 + 1] = b;
  }
  __syncthreads();

  // -- three channel mixes: Y_m[o,i] = sum_c W_m[c,o] * X[c,i], f32 WMMA --
  const int w    = t >> 5;          // wave = output-channel tile (o in [16w,16w+16))
  const int lane = t & 31;
  const int half = lane >> 4;       // A frag: VGPR0 holds K+0/K+2, VGPR1 K+1/K+3
  const int lc   = lane & 15;
  const int col  = w * 16 + lc;     // A-matrix M index = output channel
  v8f acc0 = {0,0,0,0,0,0,0,0};
  v8f acc1 = acc0, acc2 = acc0;
  for (int kc = 0; kc < 32; ++kc) {
    const int r0 = kc*4 + half*2;                   // input-channel rows
    v2f b;  b.x  = Xs[r0*16 + lc];     b.y  = Xs[(r0+1)*16 + lc];
    v2f a0; a0.x = W0[r0*128 + col];   a0.y = W0[(r0+1)*128 + col];
    v2f a1; a1.x = W1[r0*128 + col];   a1.y = W1[(r0+1)*128 + col];
    v2f a2; a2.x = W2[r0*128 + col];   a2.y = W2[(r0+1)*128 + col];
    acc0 = __builtin_amdgcn_wmma_f32_16x16x4_f32(false, a0, false, b, (short)0, acc0, false, false);
    acc1 = __builtin_amdgcn_wmma_f32_16x16x4_f32(false, a1, false, b, (short)0, acc1, false, false);
    acc2 = __builtin_amdgcn_wmma_f32_16x16x4_f32(false, a2, false, b, (short)0, acc2, false, false);
  }

  // D layout: acc[r], lanes 0-15 -> M=r, lanes 16-31 -> M=r+8; N = lane&15.
  const float scale = 0.08838834764831845f;   // 1/sqrt(128)
  #pragma unroll
  for (int r = 0; r < 8; ++r) {
    const int row = w*16 + (half ? r + 8 : r);  // channel
    A1 [lc*132 + row] = silu_fast(acc0[r] * scale);
    A1b[lc*132 + row] = silu_fast(acc1[r] * scale);
    A2b[lc*132 + row] = silu_fast(acc2[r] * scale);
  }
  __syncthreads();   // Xs fully consumed; A1/A1b/A2b visible; OUTbuf writable

  // -- sparse Wigner-3j contractions; thread -> (channel, half of k range) --
  const int c  = t & 127;
  const int kb = (t >> 7) * 8;
  #pragma unroll
  for (int q = 0; q < 8; ++q) {
    const int k = kb + q;
    OUTbuf[(c*16 + k)*3 + 0] = A1[k*132 + c];          // emit node1
    const int e0 = __builtin_amdgcn_readfirstlane(offs[k]);
    const int e1 = __builtin_amdgcn_readfirstlane(offs[k+1]);
    float acc = 0.0f;
    for (int e = e0; e < e1; ++e) {
      const int   ij = entIJ[e];
      const float wv = entW[e];
      acc += wv * A1[(ij >> 4)*132 + c] * A1b[(ij & 15)*132 + c];
    }
    B2[k*132 + c] = acc;
    OUTbuf[(c*16 + k)*3 + 1] = acc;                    // emit node2
  }
  __syncthreads();   // B2 complete
  #pragma unroll
  for (int q = 0; q < 8; ++q) {
    const int k = kb + q;
    const int e0 = __builtin_amdgcn_readfirstlane(offs[k]);
    const int e1 = __builtin_amdgcn_readfirstlane(offs[k+1]);
    float acc = 0.0f;
    for (int e = e0; e < e1; ++e) {
      const int   ij = entIJ[e];
      const float wv = entW[e];
      acc += wv * B2[(ij >> 4)*132 + c] * A2b[(ij & 15)*132 + c];
    }
    OUTbuf[(c*16 + k)*3 + 2] = acc;                    // emit node3
  }
  __syncthreads();

  // -- coalesced non-temporal writeback: 6144 floats = 1536 v4f --
  {
    v4f* dst = (v4f*)(out + (size_t)n * (NCH*DIM*3));
    const v4f* src = (const v4f*)OUTbuf;
    #pragma unroll
    for (int q = 0; q < 6; ++q)
      __builtin_nontemporal_store(src[q*256 + t], dst + q*256 + t);
  }
}

extern "C" void kernel_launch(void* const* d_in, const int* in_sizes, int n_in,
                              void* d_out, int out_size, void* d_ws, size_t ws_size,
                              hipStream_t stream) {
  (void)in_sizes; (void)n_in; (void)out_size;
  if (ws_size < (size_t)WS_FLOATS_NEEDED * 4) return;
  const float* x  = (const float*)d_in[0];
  const float* W0 = (const float*)d_in[1];
  const float* W1 = (const float*)d_in[2];
  const float* W2 = (const float*)d_in[3];
  float* out = (float*)d_out;
  float* ws  = (float*)d_ws;

  build_w3j_dense<<<16, 256, 0, stream>>>(ws);
  w3j_norms<<<1, 64, 0, stream>>>(ws);
  w3j_pack<<<1, 1, 0, stream>>>(ws);
  tp3_main<<<NODES, 256, 0, stream>>>(x, W0, W1, W2, out,
                                      (const int*)(ws + WS_OFFS),
                                      (const int*)(ws + WS_ENTIJ),
                                      ws + WS_ENTW);
}